// GraphSAGE_90907277787727
// MI455X (gfx1250) — compile-verified
//
#include <hip/hip_runtime.h>
#include <math.h>

// ---------------------------------------------------------------------------
// GraphSAGE inference, algebraically restructured:
//   mean_n0( concat(h1, mean_leaf) @ W1 )  ==  mean_n0(concat(h1, mean_leaf)) @ W1
// so the 25600x256x128 GEMM collapses to 1024x256x128. The run time is then
// bounded by the ~145 MB of random embedding-row gathers (~6.3 us @ 23.3 TB/s).
// GEMMs use native fp32 WMMA (V_WMMA_F32_16X16X4_F32) to stay precision-
// faithful to the fp32 reference.
// ---------------------------------------------------------------------------

typedef __attribute__((ext_vector_type(2))) float v2f;
typedef __attribute__((ext_vector_type(8))) float v8f;

#define BATCH 1024
#define DIM   128
#define NN0   25
#define NN1   10
#define KCAT  256   // concat feature dim (2*128)

// ---------------------------------------------------------------------------
// Kernel 1: per-batch gather + mean.
//   Xbar[b, 0:128]   = mean over 25  of embed[idx1[b,:]]
//   Xbar[b, 128:256] = mean over 250 of embed[idx2[b,:,:]]   (mean-of-means == global mean)
//   H0in[b, 0:128]   = embed[roots[b]]
// Block = 256 threads = 8 waves. Wave g handles rows j = g, g+8, ... ; the 32
// lanes of a wave load one 512-byte embedding row fully coalesced as float4.
// ---------------------------------------------------------------------------
__global__ __launch_bounds__(256) void gather_mean_kernel(
    const float* __restrict__ emb,    // [NODE_CNT+1, 128]
    const int*   __restrict__ roots,  // [B]
    const int*   __restrict__ idx1,   // [B, 25]
    const int*   __restrict__ idx2,   // [B, 250]
    float*       __restrict__ Xbar,   // [B, 256]
    float*       __restrict__ H0in)   // [B, 256]
{
    const int b   = blockIdx.x;
    const int tid = threadIdx.x;
    const int c   = tid & 31;   // float4 column within the 128-float row
    const int g   = tid >> 5;   // wave index 0..7

    const float4* emb4 = (const float4*)emb;

    __shared__ float4 s1[8][32];
    __shared__ float4 s2[8][32];

    // idx1 rows (25 of them)
    float4 a1 = make_float4(0.f, 0.f, 0.f, 0.f);
    for (int j = g; j < NN0; j += 8) {
        const int r = idx1[b * NN0 + j];
        const float4 v = emb4[(size_t)r * 32 + c];
        a1.x += v.x; a1.y += v.y; a1.z += v.z; a1.w += v.w;
    }

    // idx2 rows (250 of them)
    float4 a2 = make_float4(0.f, 0.f, 0.f, 0.f);
    for (int j = g; j < NN0 * NN1; j += 8) {
        const int r = idx2[b * (NN0 * NN1) + j];
        const float4 v = emb4[(size_t)r * 32 + c];
        a2.x += v.x; a2.y += v.y; a2.z += v.z; a2.w += v.w;
    }

    s1[g][c] = a1;
    s2[g][c] = a2;

    // Root embedding -> H0in[b, 0:128] (independent of the reduction)
    if (tid < 32) {
        const int r = roots[b];
        ((float4*)(H0in + (size_t)b * KCAT))[c] = emb4[(size_t)r * 32 + c];
    }

    __syncthreads();

    if (tid < 32) {
        float4 s = s1[0][tid];
        #pragma unroll
        for (int k = 1; k < 8; ++k) {
            const float4 v = s1[k][tid];
            s.x += v.x; s.y += v.y; s.z += v.z; s.w += v.w;
        }
        const float sc = 1.0f / (float)NN0;
        s.x *= sc; s.y *= sc; s.z *= sc; s.w *= sc;
        ((float4*)(Xbar + (size_t)b * KCAT))[tid] = s;
    } else if (tid < 64) {
        const int cc = tid - 32;
        float4 s = s2[0][cc];
        #pragma unroll
        for (int k = 1; k < 8; ++k) {
            const float4 v = s2[k][cc];
            s.x += v.x; s.y += v.y; s.z += v.z; s.w += v.w;
        }
        const float sc = 1.0f / (float)(NN0 * NN1);
        s.x *= sc; s.y *= sc; s.z *= sc; s.w *= sc;
        ((float4*)(Xbar + (size_t)b * KCAT + DIM))[cc] = s;
    }
}

// ---------------------------------------------------------------------------
// Kernel 2/3: C[M x 128] = A[M x 256] @ W[256 x 128] (+bias, sigmoid) via
// V_WMMA_F32_16X16X4_F32. One wave per 16x16 output tile; K-loop of 64 WMMAs.
//
// fp32 16x4 A fragment layout (ISA 7.12.2): lanes 0-15 hold M=lane, K={0,1};
// lanes 16-31 hold M=lane-16, K={2,3}  -> per-lane float2 from row-major A.
// fp32 4x16 B fragment mirrors it:      lanes 0-15 hold N=lane, K={0,1};
// lanes 16-31 hold N=lane-16, K={2,3}.
// C/D: VGPR r, lane l -> (M = r + 8*(l>>4), N = l&15).
// ---------------------------------------------------------------------------
template <bool SIGMOID>
__global__ __launch_bounds__(32) void wmma_gemm_kernel(
    const float* __restrict__ A,     // [M, 256]
    const float* __restrict__ W,     // [256, 128] row-major (K x N)
    const float* __restrict__ bias,  // [128] or nullptr
    float*       __restrict__ C,     // [M, ldc] (writes 16x16 tiles at col off)
    int ldc)
{
    const int tile  = blockIdx.x;
    const int tn    = tile & 7;        // N tile (128/16 = 8)
    const int tm    = tile >> 3;       // M tile
    const int lane  = threadIdx.x;

    const int M0    = tm * 16;
    const int Nc    = tn * 16;
    const int l15   = lane & 15;
    const int khalf = (lane >> 4) * 2; // 0 or 2

    const float* arow = A + (size_t)(M0 + l15) * KCAT + khalf;
    const float* wcol = W + (size_t)khalf * DIM + (Nc + l15);

    v8f acc = {0.f, 0.f, 0.f, 0.f, 0.f, 0.f, 0.f, 0.f};

    #pragma unroll 8
    for (int k = 0; k < KCAT; k += 4) {
        v2f a = *(const v2f*)(arow + k);
        v2f bf;
        bf.x = wcol[(size_t)k * DIM];
        bf.y = wcol[(size_t)(k + 1) * DIM];
        acc = __builtin_amdgcn_wmma_f32_16x16x4_f32(
            /*neg_a=*/false, a, /*neg_b=*/false, bf,
            /*c_mod=*/(short)0, acc, /*reuse_a=*/false, /*reuse_b=*/false);
    }

    const int mbase = M0 + ((lane >> 4) << 3);
    const int n     = Nc + l15;
    const float bv  = SIGMOID ? bias[n] : 0.0f;

    #pragma unroll
    for (int r = 0; r < 8; ++r) {
        float v = acc[r];
        if (SIGMOID) {
            v += bv;
            v = 1.0f / (1.0f + __expf(-v));
        }
        C[(size_t)(mbase + r) * ldc + n] = v;
    }
}

// ---------------------------------------------------------------------------
// Launcher. Inputs (setup_inputs order):
//   0: embed_table f32 [1000001,128]   1: W0 f32 [256,128]   2: b0 f32 [128]
//   3: W1 f32 [256,128]                4: roots i32 [1024]
//   5: idx1 i32 [1024,25]              6: idx2 i32 [1024,25,10]
// Output: f32 [1024,128]. Workspace: Xbar (1 MB) + H0in (1 MB).
// ---------------------------------------------------------------------------
extern "C" void kernel_launch(void* const* d_in, const int* in_sizes, int n_in,
                              void* d_out, int out_size, void* d_ws, size_t ws_size,
                              hipStream_t stream) {
    const float* emb   = (const float*)d_in[0];
    const float* W0    = (const float*)d_in[1];
    const float* b0    = (const float*)d_in[2];
    const float* W1    = (const float*)d_in[3];
    const int*   roots = (const int*)d_in[4];
    const int*   idx1  = (const int*)d_in[5];
    const int*   idx2  = (const int*)d_in[6];

    float* out  = (float*)d_out;
    float* Xbar = (float*)d_ws;                       // [1024, 256]
    float* H0in = Xbar + (size_t)BATCH * KCAT;        // [1024, 256]

    // 1) Gather + mean (memory-bound: ~145 MB of embedding rows)
    gather_mean_kernel<<<BATCH, 256, 0, stream>>>(emb, roots, idx1, idx2, Xbar, H0in);

    // 2) mean0 = Xbar @ W1  -> H0in[:, 128:256]
    wmma_gemm_kernel<false><<<(BATCH / 16) * (DIM / 16), 32, 0, stream>>>(
        Xbar, W1, nullptr, H0in + DIM, KCAT);

    // 3) out = sigmoid(H0in @ W0 + b0)
    wmma_gemm_kernel<true><<<(BATCH / 16) * (DIM / 16), 32, 0, stream>>>(
        H0in, W0, b0, out, DIM);
}